// MonotonicSpline_18829136626217
// MI455X (gfx1250) — compile-verified
//
#include <hip/hip_runtime.h>
#include <math.h>

// MonotonicSpline on gfx1250 (MI455X).
// Pure HBM-streaming problem: 4.19M elems x 8B = 33.5 MB -> ~1.44 us floor at
// 23.3 TB/s. Uniform-knot cubic B-spline => closed-form 4-tap evaluation.
// CDNA5 paths: async global->LDS B128 double-buffering with ASYNCcnt split
// waits + nontemporal B128 stores. WMMA deliberately unused (GEMV-with-gather
// would materialize a 4M x 43 basis => ~20x more traffic).

#define TPB 256
#define NKNOTS 40
#define NCOEF 43                      // NUM_KNOTS + K
#define MIN_DELTA (0.5f / 40.0f)
#define MAX_DELTA (3.0f / 40.0f)

typedef __attribute__((ext_vector_type(4))) float v4f;
typedef __attribute__((ext_vector_type(4))) int   v4i;
// AS(1) = global, AS(3) = LDS.
typedef __attribute__((address_space(1))) v4i* gptr_v4i;
typedef __attribute__((address_space(3))) v4i* lptr_v4i;

#if defined(__has_builtin)
#  if __has_builtin(__builtin_amdgcn_global_load_async_to_lds_b128) && \
      __has_builtin(__builtin_amdgcn_s_wait_asynccnt)
#    define USE_ASYNC_LDS 1
#  endif
#endif
#ifndef USE_ASYNC_LDS
#  define USE_ASYNC_LDS 0
#endif

// Closed-form uniform cubic B-spline evaluation; c is a 44-entry table
// (43 coefficients + 1 zero pad so s==40 (x==1, t==0, w3==0) stays in-bounds).
__device__ __forceinline__ float spline_eval1(float xi, const float* __restrict__ c) {
  float xn = fminf(fmaxf(xi, 0.0f), 1.0f);
  float xs = xn * (float)NKNOTS;
  int s = (int)xs;
  if (s > NKNOTS) s = NKNOTS;
  float t  = xs - (float)s;
  float t2 = t * t;
  float t3 = t2 * t;
  float u  = 1.0f - t;
  const float k6 = 1.0f / 6.0f;
  float w0 = u * u * u * k6;
  float w1 = (3.0f * t3 - 6.0f * t2 + 4.0f) * k6;
  float w2 = (-3.0f * t3 + 3.0f * t2 + 3.0f * t + 1.0f) * k6;
  float w3 = t3 * k6;
  return w0 * c[s] + w1 * c[s + 1] + w2 * c[s + 2] + w3 * c[s + 3];
}

__global__ __launch_bounds__(TPB) void MonotonicSpline_eval_kernel(
    const float* __restrict__ x, const float* __restrict__ c0,
    const float* __restrict__ raw_delta, float* __restrict__ y,
    int n, int n4, int ntiles) {
  __shared__ float s_coef[NCOEF + 1];
  __shared__ float s_tmp[NCOEF];        // per-lane cumsum contributions
#if USE_ASYNC_LDS
  __shared__ float s_buf[2][TPB * 4];   // 2 x 4KB: per-lane private staging slots
#endif

  const int lane    = (int)threadIdx.x;
  const int tstride = (int)gridDim.x;
  int ti = (int)blockIdx.x;

#if USE_ASYNC_LDS
  float* slot0 = &s_buf[0][lane * 4];
  float* slot1 = &s_buf[1][lane * 4];
  // Kick off tile 0 immediately so HBM ramps at cycle ~0; the coefficient
  // table build below overlaps the first tile's memory latency.
  if (ti < ntiles) {
    int i4 = ti * TPB + lane; if (i4 >= n4) i4 = 0;   // clamp: issued, never consumed
    __builtin_amdgcn_global_load_async_to_lds_b128(
        (gptr_v4i)(void*)(x + (size_t)i4 * 4),
        (lptr_v4i)(void*)slot0, 0, 0);
  }
#endif

  // --- Parallel coefficient table build (replaces serial thread-0 loop) ---
  // Lane i holds contribution i of cumsum([c0, deltas]); lanes run the
  // sigmoid concurrently on the VALU (v_exp_f32), then each lane does a
  // short left-to-right prefix sum (same accumulation order as jnp.cumsum).
  if (lane == 0) {
    s_tmp[0] = c0[0];
  } else if (lane < NCOEF) {
    float r  = raw_delta[lane - 1];
    float sg = 1.0f / (1.0f + __expf(-r));
    s_tmp[lane] = MIN_DELTA + (MAX_DELTA - MIN_DELTA) * sg;
  }
  if (lane == NCOEF) s_coef[NCOEF] = 0.0f;   // pad entry
  __syncthreads();
  if (lane < NCOEF) {
    float acc = 0.0f;
    for (int i = 0; i <= lane; ++i) acc += s_tmp[i];
    s_coef[lane] = acc;
  }
  __syncthreads();

#if USE_ASYNC_LDS
  // Double-buffered async global->LDS pipeline. Each lane stages its own 16B
  // (no cross-lane sharing -> per-wave ASYNCcnt wait suffices, no barriers).
  int phase = 0;
  for (; ti < ntiles; ti += tstride) {
    int tn = ti + tstride;
    if (tn < ntiles) {
      int j4 = tn * TPB + lane; if (j4 >= n4) j4 = 0;
      __builtin_amdgcn_global_load_async_to_lds_b128(
          (gptr_v4i)(void*)(x + (size_t)j4 * 4),
          (lptr_v4i)(void*)(phase ? slot0 : slot1), 0, 0);
      __builtin_amdgcn_s_wait_asynccnt(1);   // tile ti done (in-order completion)
    } else {
      __builtin_amdgcn_s_wait_asynccnt(0);   // drain for the final tile
    }
    v4f xv = *(const v4f*)(phase ? slot1 : slot0);
    v4f r;
#pragma unroll
    for (int k = 0; k < 4; ++k) r[k] = spline_eval1(xv[k], s_coef);
    int i4 = ti * TPB + lane;
    if (i4 < n4)
      __builtin_nontemporal_store(r, (v4f*)(y + (size_t)i4 * 4));  // write-once: NT
    phase ^= 1;
  }
#else
  // Fallback: direct vectorized streaming (global_load_b128 + NT store).
  for (; ti < ntiles; ti += tstride) {
    int i4 = ti * TPB + lane;
    if (i4 < n4) {
      v4f xv = *(const v4f*)(x + (size_t)i4 * 4);
      v4f r;
#pragma unroll
      for (int k = 0; k < 4; ++k) r[k] = spline_eval1(xv[k], s_coef);
      __builtin_nontemporal_store(r, (v4f*)(y + (size_t)i4 * 4));
    }
  }
#endif

  // Scalar tail (n = 4M here, so normally empty; kept for generality).
  if (blockIdx.x == 0 && threadIdx.x == 0) {
    for (int i = n4 * 4; i < n; ++i) y[i] = spline_eval1(x[i], s_coef);
  }
}

extern "C" void kernel_launch(void* const* d_in, const int* in_sizes, int n_in,
                              void* d_out, int out_size, void* d_ws, size_t ws_size,
                              hipStream_t stream) {
  // setup_inputs order: x, grid, c0, raw_delta. The grid is provably uniform
  // (h = 1/40 including extension knots), so it is folded into the closed form
  // and d_in[1] is unused.
  const float* x         = (const float*)d_in[0];
  const float* c0        = (const float*)d_in[2];
  const float* raw_delta = (const float*)d_in[3];
  float* out             = (float*)d_out;

  int n      = out_size;                 // 1024*4096 = 4,194,304
  int n4     = n >> 2;                   // float4 count
  int ntiles = (n4 + TPB - 1) / TPB;     // 4096 tiles of 4KB
  int blocks = ntiles < 1024 ? (ntiles > 0 ? ntiles : 1) : 1024;

  MonotonicSpline_eval_kernel<<<blocks, TPB, 0, stream>>>(
      x, c0, raw_delta, out, n, n4, ntiles);
}